// HarmonicEstimation_43568148251035
// MI455X (gfx1250) — compile-verified
//
#include <hip/hip_runtime.h>
#include <stdint.h>

// Problem constants (x: [16, 1, 1025, 1024] float32)
#define B_DIM 16
#define F_DIM 1025
#define T_DIM 1024
#define MAX_PEAKS 5
#define LIMIT 1020          // F - FREQ_MARGIN - 2 = 1025 - 5
#define MAX_POWER 0.1f

#define WAVES1 8            // pass-1 waves per block, each owns 128 f-bins
#define ROWS_PER_WAVE 128   // 8*128 = 1024 bins -> f = 1..1024 (bin 0 excluded)

// Dynamic LDS layout for pass 1:
//   stage : 8 waves * 128 rows * 32 floats = 131072 B  (async-load target)
//   mv    : 32 t * 8 chunks * 5 floats     =   5120 B  (partial top-5 values)
//   mi    : 32 t * 8 chunks * 5 ints       =   5120 B  (partial top-5 indices)
#define STAGE_BYTES (WAVES1 * ROWS_PER_WAVE * 32 * 4)
#define MERGE_V_OFF (STAGE_BYTES)
#define MERGE_I_OFF (STAGE_BYTES + 32 * 8 * 5 * 4)
#define SMEM_BYTES  (STAGE_BYTES + 2 * 32 * 8 * 5 * 4)

// ---------------- CDNA5 async global->LDS helpers --------------------------

#if defined(__HIP_DEVICE_COMPILE__)
#  if __has_builtin(__builtin_amdgcn_s_wait_asynccnt)
#    define WAIT_ASYNC(n) do { __builtin_amdgcn_s_wait_asynccnt(n); \
                               asm volatile("" ::: "memory"); } while (0)
#  else
#    define WAIT_ASYNC(n) asm volatile("s_wait_asynccnt " #n ::: "memory")
#  endif
#else
#  define WAIT_ASYNC(n) do { } while (0)
#endif

typedef int v4i __attribute__((ext_vector_type(4)));

__device__ __forceinline__ void async_load_b128(const float* g, void* l) {
#if defined(__HIP_DEVICE_COMPILE__)
#if __has_builtin(__builtin_amdgcn_global_load_async_to_lds_b128)
  // Signature (probe-confirmed by diagnostic): (v4i AS1*, v4i AS3*, Ii, Ii)
  __builtin_amdgcn_global_load_async_to_lds_b128(
      (__attribute__((address_space(1))) v4i*)(void*)g,
      (__attribute__((address_space(3))) v4i*)l,
      /*offset=*/0, /*cpol=*/0);
#else
  unsigned lo = (unsigned)(unsigned long long)
      (__attribute__((address_space(3))) char*)l;
  asm volatile("global_load_async_to_lds_b128 %0, %1, off"
               :: "v"(lo), "v"((const void*)g) : "memory");
#endif
#else
  (void)g; (void)l;
#endif
}

// ---------------- Pass 1: per-column f0 extraction -------------------------

#define SWAP5(a, b, ai, bi) { float tv = a; a = b; b = tv; \
                              int   ti = ai; ai = bi; bi = ti; }
#define PROC_ROW(r) do {                                                     \
    float val = swave[(r) * 32 + l];                                         \
    if (val > v4) {                                                          \
      v4 = val; i4 = fbase + (r);                                            \
      if (v4 > v3) { SWAP5(v3, v4, i3, i4);                                  \
        if (v3 > v2) { SWAP5(v2, v3, i2, i3);                                \
          if (v2 > v1) { SWAP5(v1, v2, i1, i2);                              \
            if (v1 > v0) { SWAP5(v0, v1, i0, i1); } } } }                    \
    } } while (0)

__global__ void __launch_bounds__(256)
f0_extract_kernel(const float* __restrict__ x, int2* __restrict__ tbl) {
  extern __shared__ char smem[];
  float* stage = (float*)smem;
  float* mv    = (float*)(smem + MERGE_V_OFF);
  int*   mi    = (int*)  (smem + MERGE_I_OFF);

  const int bb = blockIdx.x >> 5;          // 512 blocks = 16 b * 32 t-tiles
  const int t0 = (blockIdx.x & 31) * 32;
  const int w  = threadIdx.x >> 5;         // wave id (wave32)
  const int l  = threadIdx.x & 31;         // lane = t within tile

  const int fbase = 1 + w * ROWS_PER_WAVE; // bin 0 excluded from peak search
  const float* gwave =
      x + (size_t)bb * F_DIM * T_DIM + (size_t)fbase * T_DIM + t0;
  float* swave = stage + w * (ROWS_PER_WAVE * 32);

  // Issue all 32 async b128 loads: op j covers rows [4j, 4j+4).
  // lane -> (row sub-index, 16B t sub-offset); fully coalesced 512B/op.
  const int rsub = l >> 3;
  const int tsub = (l & 7) * 4;
  for (int j = 0; j < 32; ++j) {
    const int r = j * 4 + rsub;
    async_load_b128(gwave + (size_t)r * T_DIM + tsub, swave + r * 32 + tsub);
  }

  // Running top-5 (descending). Strict '>' keeps smaller index on ties,
  // matching iterative max+mask-out top_k semantics.
  float v0 = -1e30f, v1 = -1e30f, v2 = -1e30f, v3 = -1e30f, v4 = -1e30f;
  int   i0 = 0, i1 = 0, i2 = 0, i3 = 0, i4 = 0;

  // Async loads complete in order: <=24 outstanding => first 8 ops (rows
  // 0..31) landed, etc. Constant-immediate waits.
  WAIT_ASYNC(24);
  for (int r = 0;  r < 32;  ++r) PROC_ROW(r);
  WAIT_ASYNC(16);
  for (int r = 32; r < 64;  ++r) PROC_ROW(r);
  WAIT_ASYNC(8);
  for (int r = 64; r < 96;  ++r) PROC_ROW(r);
  WAIT_ASYNC(0);
  for (int r = 96; r < 128; ++r) PROC_ROW(r);

  // Publish partial top-5 for (t=l, chunk=w).
  const int mb = (l * 8 + w) * 5;
  mv[mb + 0] = v0; mv[mb + 1] = v1; mv[mb + 2] = v2;
  mv[mb + 3] = v3; mv[mb + 4] = v4;
  mi[mb + 0] = i0; mi[mb + 1] = i1; mi[mb + 2] = i2;
  mi[mb + 3] = i3; mi[mb + 4] = i4;
  __syncthreads();

  // One thread per t merges the 8 partial lists (40 candidates) and selects
  // the global top-5 via 5x argmax with smaller-index tie-break.
  if (threadIdx.x < 32) {
    const int t = threadIdx.x;
    float* lv = mv + t * 40;
    int*   li = mi + t * 40;
    int  minIdx = 0x7fffffff;
    bool any = false;
    for (int p = 0; p < MAX_PEAKS; ++p) {
      float bv = -1e30f; int bi = 0x7fffffff; int bj = 0;
      for (int j = 0; j < 40; ++j) {
        const float vv = lv[j];
        const int   ii = li[j];
        if ((vv > bv) || (vv == bv && ii < bi)) { bv = vv; bi = ii; bj = j; }
      }
      lv[bj] = -1e30f;                 // mask out
      if (bv > MAX_POWER) { any = true; minIdx = min(minIdx, bi); }
    }
    const int f0   = any ? minIdx : 0;
    const int qlim = (f0 > 0) ? (LIMIT / f0) : 0;
    tbl[bb * T_DIM + t0 + t] = make_int2(f0, qlim * f0);
  }
}

// ---------------- Pass 2: harmonic comb mask (write-only stream) -----------
//
// Reference per bin k: i_last = min((k+3)/f0, 1020/f0) * f0;
//   out = (f0>0 && i_last>=f0 && i_last>=k-3) ? max(1-0.5*|k-i_last|/3, 0.5)
//                                             : 0.5
// i_last < k-3 implies dist>3 implies value 0.5 anyway, so that check is
// subsumed by the clamp. Since f0 is fixed per lane, i_last is computed by
// marching: it advances by f0 exactly when k+3 reaches the next multiple
// (and stays <= ilmax). One real division only at loop entry.

__global__ void __launch_bounds__(512)
mask_kernel(const int2* __restrict__ tbl, float* __restrict__ out) {
  const int bb = blockIdx.x >> 5;
  const int t0 = (blockIdx.x & 31) * 32;
  const int w  = threadIdx.x >> 5;   // 16 waves split F
  const int l  = threadIdx.x & 31;
  const int t  = t0 + l;
  const int fs = (F_DIM * w) >> 4;
  const int fe = (F_DIM * (w + 1)) >> 4;

  const int2 c     = tbl[bb * T_DIM + t];
  const int  f0    = c.x;
  const int  ilmax = c.y;            // (1020/f0)*f0
  float* o = out + (size_t)bb * F_DIM * T_DIM + t;

  if (f0 <= 0 || ilmax < f0) {       // no f0, or f0 > 1020: mask is all 0.5
    for (int k = fs; k < fe; ++k)
      __builtin_nontemporal_store(0.5f, o + (size_t)k * T_DIM);
    return;
  }

  // i_last state for k = fs-1 (loop advances it for each k >= fs).
  int il = ((fs + 2) / f0) * f0;
  if (il > ilmax) il = ilmax;

  for (int k = fs; k < fe; ++k) {
    const int iln = il + f0;
    if ((k + 3 >= iln) && (iln <= ilmax)) il = iln;
    int dist = k - il; dist = (dist < 0) ? -dist : dist;
    float val = 1.0f - (0.5f * (float)dist) / 3.0f;
    val = fmaxf(val, 0.5f);
    val = (il > 0) ? val : 0.5f;     // i_last >= f0 condition
    __builtin_nontemporal_store(val, o + (size_t)k * T_DIM);
  }
}

// ---------------- launch ---------------------------------------------------

extern "C" void kernel_launch(void* const* d_in, const int* in_sizes, int n_in,
                              void* d_out, int out_size, void* d_ws,
                              size_t ws_size, hipStream_t stream) {
  (void)in_sizes; (void)n_in; (void)out_size; (void)ws_size;
  const float* x   = (const float*)d_in[0];
  float*       out = (float*)d_out;
  int2*        tbl = (int2*)d_ws;    // 16384 * 8 B = 128 KB scratch

  // 141312 B dynamic LDS per block (2 blocks/WGP within 320 KB).
  (void)hipFuncSetAttribute((const void*)f0_extract_kernel,
                            hipFuncAttributeMaxDynamicSharedMemorySize,
                            SMEM_BYTES);

  f0_extract_kernel<<<dim3(B_DIM * (T_DIM / 32)), dim3(256), SMEM_BYTES,
                      stream>>>(x, tbl);
  mask_kernel<<<dim3(B_DIM * (T_DIM / 32)), dim3(512), 0, stream>>>(tbl, out);
}